// ChamferDistanceLoss_40939628265652
// MI455X (gfx1250) — compile-verified
//
#include <hip/hip_runtime.h>

typedef __attribute__((ext_vector_type(2))) float v2f;
typedef __attribute__((ext_vector_type(8))) float v8f;
typedef __attribute__((ext_vector_type(4))) int v4i;
typedef __attribute__((address_space(1))) v4i gv4i;   // global int4
typedef __attribute__((address_space(3))) v4i lv4i;   // LDS int4

#define CB 4
#define CN 8192
#define CM 8192
#define MSPLIT 8
#define MCHUNK (CM / MSPLIT)                    // 1024 columns per workgroup
#define ROWS_PER_WAVE 32
#define WAVES 8
#define ROWS_PER_BLOCK (ROWS_PER_WAVE * WAVES)  // 256
#define ROWBLKS (CN / ROWS_PER_BLOCK)           // 32
#define INF_BITS 0x7F800000u

// ---------------- init: +inf-fill the min arrays ----------------
__global__ void chamfer_init(unsigned int* __restrict__ rowmin,
                             unsigned int* __restrict__ colmin) {
  int i = blockIdx.x * blockDim.x + threadIdx.x;
  if (i < CB * CN) rowmin[i] = INF_BITS;
  if (i < CB * CM) colmin[i] = INF_BITS;
}

// ---------------- main: WMMA-tiled pairwise min-d^2 ----------------
__global__ void __launch_bounds__(256) chamfer_main(
    const float* __restrict__ pred, const float* __restrict__ gt,
    unsigned int* __restrict__ rowmin, unsigned int* __restrict__ colmin) {
  __shared__ __align__(16) float lds_gt[MCHUNK * 3];   // 12 KB staged gt chunk
  __shared__ unsigned int lds_colmin[MCHUNK];          // 4 KB column minima

  const int wg     = blockIdx.x;
  const int mchunk = wg % MSPLIT;
  const int rowblk = (wg / MSPLIT) % ROWBLKS;
  const int b      = wg / (MSPLIT * ROWBLKS);

  const int tid    = threadIdx.x;
  const int wave   = tid >> 5;
  const int lane   = tid & 31;
  const int lane16 = lane & 15;
  const bool hi    = lane >= 16;

  const float* P = pred + (size_t)b * CN * 3;
  const float* G = gt   + (size_t)b * CM * 3;
  const int mbeg = mchunk * MCHUNK;
  const float* Gc = G + (size_t)mbeg * 3;              // 16B-aligned chunk base

  for (int i = tid; i < MCHUNK; i += 256) lds_colmin[i] = INF_BITS;

  // ---- stage gt chunk into LDS via async copies (ASYNCcnt path) ----
#if __has_builtin(__builtin_amdgcn_global_load_async_to_lds_b128)
  {
    gv4i* gsrc = (gv4i*)Gc;                             // cast away const, AS1 int4*
    lv4i* ldst = (lv4i*)lds_gt;                         // AS3 int4*
#pragma unroll
    for (int k = 0; k < 3; ++k) {                       // 768 x B128 = 12 KB
      int idx = tid + k * 256;
      __builtin_amdgcn_global_load_async_to_lds_b128(gsrc + idx, ldst + idx, 0, 0);
    }
  }
#if __has_builtin(__builtin_amdgcn_s_wait_asynccnt)
  __builtin_amdgcn_s_wait_asynccnt(0);
#else
  asm volatile("s_wait_asynccnt 0x0" ::: "memory");
#endif
#else
  for (int i = tid; i < MCHUNK * 3; i += 256) lds_gt[i] = Gc[i];
#endif
  __syncthreads();

  const int n0 = rowblk * ROWS_PER_BLOCK + wave * ROWS_PER_WAVE;

  // A tiles: rows n0..n0+15 and n0+16..n0+31. Lane L holds row L%16.
  // A f32 16x4 layout: VGPR0 = K{0|2}, VGPR1 = K{1|3} (lo|hi lane half).
  // Embed row as (px,py,pz,1) so B can carry (-2g, g^2).
  float p2v0[8], p2v1[8];
  v2f a0, a1;
  {
    const float* p = P + (size_t)(n0 + lane16) * 3;
    float px = p[0], py = p[1], pz = p[2];
    float p2 = px * px + py * py + pz * pz;
    a0.x = hi ? pz : px;
    a0.y = hi ? 1.0f : py;
#pragma unroll
    for (int v = 0; v < 8; ++v) p2v0[v] = __shfl(p2, v + (hi ? 8 : 0), 32);

    const float* q = P + (size_t)(n0 + 16 + lane16) * 3;
    float qx = q[0], qy = q[1], qz = q[2];
    float q2 = qx * qx + qy * qy + qz * qz;
    a1.x = hi ? qz : qx;
    a1.y = hi ? 1.0f : qy;
#pragma unroll
    for (int v = 0; v < 8; ++v) p2v1[v] = __shfl(q2, v + (hi ? 8 : 0), 32);
  }

  float rmin0[8], rmin1[8];
#pragma unroll
  for (int v = 0; v < 8; ++v) { rmin0[v] = __builtin_inff(); rmin1[v] = __builtin_inff(); }

  for (int mt = 0; mt < MCHUNK; mt += 16) {
    const float* g = &lds_gt[(mt + lane16) * 3];        // broadcast ds_load
    float gx = g[0], gy = g[1], gz = g[2];
    float g2 = gx * gx + gy * gy + gz * gz;
    // B col = (-2gx,-2gy,-2gz, g^2), same slot->K mapping as A.
    v2f bm;
    bm.x = hi ? (-2.0f * gz) : (-2.0f * gx);
    bm.y = hi ? g2 : (-2.0f * gy);

    v8f c0 = {}, c1 = {};
    // acc = g^2 - 2 p.g  per (row, col) element
    c0 = __builtin_amdgcn_wmma_f32_16x16x4_f32(false, a0, false, bm,
                                               (short)0, c0, false, false);
    c1 = __builtin_amdgcn_wmma_f32_16x16x4_f32(false, a1, false, bm,
                                               (short)0, c1, false, false);

    float cmin = __builtin_inff();
#pragma unroll
    for (int v = 0; v < 8; ++v) {
      // clamp >=0 so uint bit-pattern ordering is monotone; sqrt deferred
      float d0 = fmaxf(p2v0[v] + c0[v], 0.0f);
      float d1 = fmaxf(p2v1[v] + c1[v], 0.0f);
      rmin0[v] = fminf(rmin0[v], d0);
      rmin1[v] = fminf(rmin1[v], d1);
      cmin = fminf(cmin, fminf(d0, d1));
    }
    // combine the two lane halves (rows v and v+8 cover same column set)
    cmin = fminf(cmin, __shfl_xor(cmin, 16, 32));
    if (!hi) atomicMin(&lds_colmin[mt + lane16], __float_as_uint(cmin));
  }

  // Row minima: reduce over the 16 column-class lanes inside each half.
#pragma unroll
  for (int v = 0; v < 8; ++v) {
    float r0 = rmin0[v], r1 = rmin1[v];
#pragma unroll
    for (int s = 1; s < 16; s <<= 1) {
      r0 = fminf(r0, __shfl_xor(r0, s, 32));
      r1 = fminf(r1, __shfl_xor(r1, s, 32));
    }
    if (lane16 == 0) {
      int row0 = n0 + v + (hi ? 8 : 0);
      atomicMin(&rowmin[(size_t)b * CN + row0],      __float_as_uint(r0));
      atomicMin(&rowmin[(size_t)b * CN + row0 + 16], __float_as_uint(r1));
    }
  }

  __syncthreads();
  for (int i = tid; i < MCHUNK; i += 256)
    atomicMin(&colmin[(size_t)b * CM + mbeg + i], lds_colmin[i]);
}

// ---------------- reduce: sqrt + deterministic tree sum ----------------
__global__ void __launch_bounds__(256) chamfer_reduce(
    const unsigned int* __restrict__ rowmin, const unsigned int* __restrict__ colmin,
    float* __restrict__ out) {
  __shared__ float sdata[256];
  int tid = threadIdx.x;
  float s1 = 0.f, s2 = 0.f;
  for (int i = tid; i < CB * CN; i += 256) s1 += sqrtf(__uint_as_float(rowmin[i]));
  for (int i = tid; i < CB * CM; i += 256) s2 += sqrtf(__uint_as_float(colmin[i]));
  sdata[tid] = s1 * (1.0f / (CB * CN)) + s2 * (1.0f / (CB * CM));
  __syncthreads();
  for (int s = 128; s > 0; s >>= 1) {
    if (tid < s) sdata[tid] += sdata[tid + s];
    __syncthreads();
  }
  if (tid == 0) out[0] = sdata[0];
}

extern "C" void kernel_launch(void* const* d_in, const int* in_sizes, int n_in,
                              void* d_out, int out_size, void* d_ws, size_t ws_size,
                              hipStream_t stream) {
  (void)in_sizes; (void)n_in; (void)out_size; (void)ws_size;
  const float* pred = (const float*)d_in[0];   // [B,N,3] f32
  const float* gt   = (const float*)d_in[1];   // [B,M,3] f32
  float* out = (float*)d_out;                  // scalar f32

  unsigned int* rowmin = (unsigned int*)d_ws;          // B*N uints (min d^2 bits)
  unsigned int* colmin = rowmin + (size_t)CB * CN;     // B*M uints

  chamfer_init<<<(CB * CN + 255) / 256, 256, 0, stream>>>(rowmin, colmin);
  chamfer_main<<<CB * ROWBLKS * MSPLIT, 256, 0, stream>>>(pred, gt, rowmin, colmin);
  chamfer_reduce<<<1, 256, 0, stream>>>(rowmin, colmin, out);
}